// GeometricLoss_17540646437328
// MI455X (gfx1250) — compile-verified
//
#include <hip/hip_runtime.h>
#include <float.h>

typedef __attribute__((ext_vector_type(2))) float v2f;
typedef __attribute__((ext_vector_type(8))) float v8f;

#define WAVES_PER_BLOCK 8
#define BLOCK_THREADS (WAVES_PER_BLOCK * 32)

__global__ void zero_out_kernel(float* out) {
    if (threadIdx.x < 3) out[threadIdx.x] = 0.0f;
}

// fast reciprocal: v_rcp_f32 + one Newton-Raphson step (~0.5 ulp)
__device__ __forceinline__ float fast_rcp(float x) {
    float r = __builtin_amdgcn_rcpf(x);
    return r * (2.0f - x * r);
}

__global__ __launch_bounds__(BLOCK_THREADS)
void geo_loss_kernel(const float* __restrict__ pred_x,
                     const float* __restrict__ pred_q,
                     const float* __restrict__ target_x,
                     const float* __restrict__ target_q,
                     const float* __restrict__ c_R_w,
                     const float* __restrict__ w_P,
                     float* __restrict__ out,
                     int N, float invB)
{
    // A-matrix (6x4) for this sample: rows 0-2 = [c_R_w | -c_R_w*t],
    // rows 3-5 = [pred_R | -pred_R*pred_x]. Rows 6-15 implicit zero.
    __shared__ float ldsA[6][4];
    __shared__ float wsum[WAVES_PER_BLOCK];

    const int b = blockIdx.x;

    if (threadIdx.x == 0) {
        // normalize + conjugate quaternion
        float qw = pred_q[4*b+0], qx = pred_q[4*b+1];
        float qy = pred_q[4*b+2], qz = pred_q[4*b+3];
        float inv = rsqrtf(qw*qw + qx*qx + qy*qy + qz*qz);
        float w = qw*inv, x = -qx*inv, y = -qy*inv, z = -qz*inv;

        float r00 = 1.f - 2.f*(y*y + z*z);
        float r01 = 2.f*(x*y - z*w);
        float r02 = 2.f*(x*z + y*w);
        float r10 = 2.f*(x*y + z*w);
        float r11 = 1.f - 2.f*(x*x + z*z);
        float r12 = 2.f*(y*z - x*w);
        float r20 = 2.f*(x*z - y*w);
        float r21 = 2.f*(y*z + x*w);
        float r22 = 1.f - 2.f*(x*x + y*y);

        float tx = target_x[3*b+0], ty = target_x[3*b+1], tz = target_x[3*b+2];
        float px = pred_x[3*b+0],  py = pred_x[3*b+1],  pz = pred_x[3*b+2];

        float c00 = c_R_w[9*b+0], c01 = c_R_w[9*b+1], c02 = c_R_w[9*b+2];
        float c10 = c_R_w[9*b+3], c11 = c_R_w[9*b+4], c12 = c_R_w[9*b+5];
        float c20 = c_R_w[9*b+6], c21 = c_R_w[9*b+7], c22 = c_R_w[9*b+8];

        ldsA[0][0] = c00; ldsA[0][1] = c01; ldsA[0][2] = c02;
        ldsA[0][3] = -(c00*tx + c01*ty + c02*tz);
        ldsA[1][0] = c10; ldsA[1][1] = c11; ldsA[1][2] = c12;
        ldsA[1][3] = -(c10*tx + c11*ty + c12*tz);
        ldsA[2][0] = c20; ldsA[2][1] = c21; ldsA[2][2] = c22;
        ldsA[2][3] = -(c20*tx + c21*ty + c22*tz);
        ldsA[3][0] = r00; ldsA[3][1] = r01; ldsA[3][2] = r02;
        ldsA[3][3] = -(r00*px + r01*py + r02*pz);
        ldsA[4][0] = r10; ldsA[4][1] = r11; ldsA[4][2] = r12;
        ldsA[4][3] = -(r10*px + r11*py + r12*pz);
        ldsA[5][0] = r20; ldsA[5][1] = r21; ldsA[5][2] = r22;
        ldsA[5][3] = -(r20*px + r21*py + r22*pz);

        // scalar losses (mean over B folded in via invB)
        float dx = px - tx, dy = py - ty, dz = pz - tz;
        float lx = sqrtf(dx*dx + dy*dy + dz*dz);
        float d0 = w - target_q[4*b+0], d1 = x - target_q[4*b+1];
        float d2 = y - target_q[4*b+2], d3 = z - target_q[4*b+3];
        float lq = sqrtf(d0*d0 + d1*d1 + d2*d2 + d3*d3);
        atomicAdd(&out[1], lx * invB);
        atomicAdd(&out[2], lq * invB);
    }
    __syncthreads();

    const int lane = threadIdx.x & 31;
    const int wave = threadIdx.x >> 5;
    const int row  = lane & 15;
    const bool hi  = lane >= 16;

    // A operand: lanes 0-15 hold A[row][0],A[row][1]; lanes 16-31 hold A[row][2],A[row][3]
    v2f Am;
    Am[0] = (row < 6) ? ldsA[row][hi ? 2 : 0] : 0.0f;
    Am[1] = (row < 6) ? ldsA[row][hi ? 3 : 1] : 0.0f;

    float acc = 0.0f;
    const float* baseP = w_P + (size_t)b * (size_t)N * 3u;
    const int c0_off = hi ? 2 : 0;   // x (lanes 0-15) / z (lanes 16-31)

    #pragma unroll 2
    for (int n0 = wave * 16; n0 < N; n0 += WAVES_PER_BLOCK * 16) {
        const float* p = baseP + (size_t)(n0 + row) * 3u;
        // prefetch next tile for this wave into near cache
        __builtin_prefetch(p + 3 * WAVES_PER_BLOCK * 16, 0, 3);

        // B operand 4x16: VGPR0 = rows {0,2}, VGPR1 = rows {1,3}
        // Uniform loads in all lanes (no EXEC dance); hi lanes select 1.0.
        float c0 = p[c0_off];
        float c1 = p[1];                 // same cachelines for hi lanes, no extra traffic
        v2f Bm;
        Bm[0] = c0;
        Bm[1] = hi ? 1.0f : c1;

        v8f C = {0.f, 0.f, 0.f, 0.f, 0.f, 0.f, 0.f, 0.f};
        v8f D = __builtin_amdgcn_wmma_f32_16x16x4_f32(
            /*neg_a=*/false, Am, /*neg_b=*/false, Bm,
            /*c_mod=*/(short)0, C, /*reuse_a=*/false, /*reuse_b=*/false);

        // lanes 0-15: D[0..2] = target_c_p(xyz), D[3..5] = pred_c_p(xyz) for point n0+lane
        float it = fast_rcp(D[2]);
        float ip = fast_rcp(D[5]);
        float l1 = fabsf(D[3]*ip - D[0]*it) + fabsf(D[4]*ip - D[1]*it);
        if (!hi) acc += l1;   // upper half of D is the zero rows -> garbage, mask out
    }

    // wave32 butterfly reduction (hi lanes contribute 0)
    for (int off = 16; off > 0; off >>= 1)
        acc += __shfl_xor(acc, off, 32);
    if (lane == 0) wsum[wave] = acc;
    __syncthreads();

    if (threadIdx.x == 0) {
        float s = 0.0f;
        for (int i = 0; i < WAVES_PER_BLOCK; ++i) s += wsum[i];
        float mean = s / (float)N;
        // jnp.nan_to_num semantics
        if (mean != mean) mean = 0.0f;
        else if (mean > FLT_MAX) mean = FLT_MAX;
        else if (mean < -FLT_MAX) mean = -FLT_MAX;
        atomicAdd(&out[0], mean * invB);
    }
}

extern "C" void kernel_launch(void* const* d_in, const int* in_sizes, int n_in,
                              void* d_out, int out_size, void* d_ws, size_t ws_size,
                              hipStream_t stream) {
    const float* pred_x   = (const float*)d_in[0];
    const float* pred_q   = (const float*)d_in[1];
    const float* target_x = (const float*)d_in[2];
    const float* target_q = (const float*)d_in[3];
    const float* c_R_w    = (const float*)d_in[4];
    const float* w_P      = (const float*)d_in[5];
    float* out = (float*)d_out;

    const int B = in_sizes[0] / 3;                 // 8192
    const int N = in_sizes[5] / (B * 3);           // 2048

    zero_out_kernel<<<1, 32, 0, stream>>>(out);
    geo_loss_kernel<<<B, BLOCK_THREADS, 0, stream>>>(
        pred_x, pred_q, target_x, target_q, c_R_w, w_P, out, N, 1.0f / (float)B);
}